// GCN_16793322127453
// MI455X (gfx1250) — compile-verified
//
#include <hip/hip_runtime.h>
#include <hip/hip_bf16.h>
#include <stdint.h>

typedef __attribute__((ext_vector_type(2))) float v2f;
typedef __attribute__((ext_vector_type(8))) float v8f;

// ---------------- CSR construction (transposed graph: incoming edges per node) ----------------

__global__ void zero_int_kernel(int* __restrict__ p, int N) {
    int i = blockIdx.x * blockDim.x + threadIdx.x;
    if (i < N) p[i] = 0;
}

__global__ void count_kernel(const int* __restrict__ dst, int* __restrict__ counts, int E) {
    int e = blockIdx.x * blockDim.x + threadIdx.x;
    if (e < E) atomicAdd(&counts[dst[e]], 1);
}

// dinv[i] = 1/sqrt(indeg + 1)   (self-loop included; exact integer degree)
__global__ void dinv_kernel(const int* __restrict__ counts, float* __restrict__ dinv, int N) {
    int i = blockIdx.x * blockDim.x + threadIdx.x;
    if (i < N) dinv[i] = 1.0f / sqrtf((float)(counts[i] + 1));
}

// Single-workgroup exclusive scan: rowptr[0..N], rowptr[N] = total
__global__ __launch_bounds__(1024) void scan_kernel(const int* __restrict__ counts,
                                                    int* __restrict__ rowptr, int N) {
    __shared__ int sdata[1024];
    __shared__ int running;
    int t = threadIdx.x;
    if (t == 0) running = 0;
    __syncthreads();
    for (int base = 0; base < N; base += 1024) {
        int v = (base + t < N) ? counts[base + t] : 0;
        sdata[t] = v;
        __syncthreads();
        for (int off = 1; off < 1024; off <<= 1) {
            int add = (t >= off) ? sdata[t - off] : 0;
            __syncthreads();
            sdata[t] += add;
            __syncthreads();
        }
        if (base + t < N) rowptr[base + t] = running + sdata[t] - v;
        __syncthreads();
        if (t == 0) running += sdata[1023];
        __syncthreads();
    }
    if (t == 0) rowptr[N] = running;
}

__global__ void copy_int_kernel(const int* __restrict__ in, int* __restrict__ out, int N) {
    int i = blockIdx.x * blockDim.x + threadIdx.x;
    if (i < N) out[i] = in[i];
}

__global__ void fill_csr_kernel(const int* __restrict__ src, const int* __restrict__ dst,
                                int* __restrict__ pos, int* __restrict__ csr, int E) {
    int e = blockIdx.x * blockDim.x + threadIdx.x;
    if (e < E) {
        int p = atomicAdd(&pos[dst[e]], 1);
        csr[p] = src[e];
    }
}

// Sort each bucket ascending: makes summation order a pure function of the input
// (atomic fill order no longer matters; duplicate srcs are identical values).
__global__ void sort_buckets_kernel(const int* __restrict__ rowptr, int* __restrict__ csr, int N) {
    int i = blockIdx.x * blockDim.x + threadIdx.x;
    if (i >= N) return;
    int s = rowptr[i], t = rowptr[i + 1];
    for (int a = s + 1; a < t; ++a) {
        int key = csr[a];
        int b = a - 1;
        while (b >= s && csr[b] > key) { csr[b + 1] = csr[b]; --b; }
        csr[b + 1] = key;
    }
}

// ---------------- WMMA GEMM with dinv-scaled epilogue ----------------
// H[i,:] = dinv[i] * (X[i,:] @ W)     X: MxK row-major, W: KxNout row-major

template <int BN, int NW, int CPW>
__global__ __launch_bounds__(NW * 32)
void gemm_scale_kernel(const float* __restrict__ X, const float* __restrict__ W,
                       const float* __restrict__ dinv, float* __restrict__ H,
                       int M, int K, int Nout) {
    constexpr int BM = 64, KB = 32;
    constexpr int ASTR = KB + 4;   // padded stride: 16B-aligned rows, bank-conflict-free frags
    constexpr int BSTR = BN + 4;

    __shared__ __align__(16) float As[BM * ASTR];
    __shared__ __align__(16) float Bs[KB * BSTR];

    const int tid  = threadIdx.x;
    const int wave = tid >> 5;
    const int lane = tid & 31;
    const int l16  = lane & 15;
    const int kkof = (lane >> 4) * 2;
    const int half = lane >> 4;

    const int rowBase = blockIdx.x * BM;
    const int nBase   = blockIdx.y * BN;
    const int twRow = (wave & 3) * 16;
    const int twCol = (wave >> 2) * (CPW * 16);

    v8f acc[CPW];
    {
        v8f z;
        for (int j = 0; j < 8; ++j) z[j] = 0.0f;
        for (int c = 0; c < CPW; ++c) acc[c] = z;
    }

    const int nK = K / KB;
    for (int kb = 0; kb < nK; ++kb) {
        const int kBase = kb * KB;
        {   // A tile: BM x KB (zero-padded beyond M)
            constexpr int SLOTS = BM * KB / 4;
            for (int s = tid; s < SLOTS; s += NW * 32) {
                int r  = s / (KB / 4);
                int c4 = s % (KB / 4);
                int grow = rowBase + r;
                float4 v = make_float4(0.f, 0.f, 0.f, 0.f);
                if (grow < M)
                    v = *(const float4*)&X[(size_t)grow * K + kBase + c4 * 4];
                *(float4*)&As[r * ASTR + c4 * 4] = v;
            }
        }
        {   // B tile: KB x BN
            constexpr int SLOTS = KB * BN / 4;
            for (int s = tid; s < SLOTS; s += NW * 32) {
                int r  = s / (BN / 4);
                int c4 = s % (BN / 4);
                float4 v = *(const float4*)&W[(size_t)(kBase + r) * Nout + nBase + c4 * 4];
                *(float4*)&Bs[r * BSTR + c4 * 4] = v;
            }
        }
        __syncthreads();

        for (int k0 = 0; k0 < KB; k0 += 4) {
            v2f a;
            {
                const float2 t2 = *(const float2*)&As[(twRow + l16) * ASTR + k0 + kkof];
                a.x = t2.x;
                a.y = t2.y;
            }
            for (int c = 0; c < CPW; ++c) {
                v2f b;
                b.x = Bs[(k0 + kkof)     * BSTR + twCol + c * 16 + l16];
                b.y = Bs[(k0 + kkof + 1) * BSTR + twCol + c * 16 + l16];
                acc[c] = __builtin_amdgcn_wmma_f32_16x16x4_f32(
                    false, a, false, b, (short)0, acc[c], false, false);
            }
        }
        __syncthreads();
    }

    for (int c = 0; c < CPW; ++c) {
        for (int g = 0; g < 8; ++g) {
            int row = rowBase + twRow + g + half * 8;
            int col = nBase + twCol + c * 16 + l16;
            if (row < M)
                H[(size_t)row * Nout + col] = acc[c][g] * dinv[row];
        }
    }
}

// ---------------- fused CSR aggregation + bias + activation ----------------
// out[i] = act( dinv[i] * (H[i] + sum_{u in csr[i]} H[u]) + bias )

template <bool RELU, int F>
__global__ void agg_csr_kernel(const float* __restrict__ H, const int* __restrict__ rowptr,
                               const int* __restrict__ csr, const float* __restrict__ dinv,
                               const float* __restrict__ bias, float* __restrict__ out, int N) {
    constexpr int G = F / 4;  // threads per node
    long long gid = (long long)blockIdx.x * blockDim.x + threadIdx.x;
    int node = (int)(gid / G);
    int c    = (int)(gid % G);
    if (node >= N) return;

    int s = rowptr[node], e = rowptr[node + 1];
    float4 a = *(const float4*)&H[(size_t)node * F + c * 4];  // self-loop term
    int u_next = (s < e) ? csr[s] : 0;
    for (int j = s; j < e; ++j) {
        int u = u_next;
        if (j + 1 < e) {
            u_next = csr[j + 1];
            __builtin_prefetch(&H[(size_t)u_next * F + c * 4], 0, 1);
        }
        float4 v = *(const float4*)&H[(size_t)u * F + c * 4];
        a.x += v.x; a.y += v.y; a.z += v.z; a.w += v.w;
    }
    float di = dinv[node];
    float4 bb = *(const float4*)&bias[c * 4];
    float4 o;
    o.x = di * a.x + bb.x;
    o.y = di * a.y + bb.y;
    o.z = di * a.z + bb.z;
    o.w = di * a.w + bb.w;
    if (RELU) {
        o.x = fmaxf(o.x, 0.f); o.y = fmaxf(o.y, 0.f);
        o.z = fmaxf(o.z, 0.f); o.w = fmaxf(o.w, 0.f);
    }
    *(float4*)&out[(size_t)node * F + c * 4] = o;
}

// ---------------- host launcher ----------------

static inline int cdiv(long long a, int b) { return (int)((a + b - 1) / b); }

extern "C" void kernel_launch(void* const* d_in, const int* in_sizes, int n_in,
                              void* d_out, int out_size, void* d_ws, size_t ws_size,
                              hipStream_t stream) {
    const float* x  = (const float*)d_in[0];
    const int*   ei = (const int*)d_in[1];
    const float* W1 = (const float*)d_in[2];
    const float* b1 = (const float*)d_in[3];
    const float* W2 = (const float*)d_in[4];
    const float* b2 = (const float*)d_in[5];
    const float* W3 = (const float*)d_in[6];
    const float* b3 = (const float*)d_in[7];

    const int N = in_sizes[0] / 512;
    const int E = in_sizes[1] / 2;
    const int* src = ei;
    const int* dst = ei + E;

    // workspace layout
    float* dinv  = (float*)d_ws;                    // N floats
    int* rowptr  = (int*)(dinv + N);                // N+1 ints
    int* pos     = rowptr + (N + 1);                // N ints (also used as counts)
    int* csr     = pos + N;                         // E ints
    uintptr_t p  = ((uintptr_t)(csr + E) + 15) & ~(uintptr_t)15;
    float* bufH  = (float*)p;                       // N*256 floats (transform outputs)
    float* bufG  = bufH + (size_t)N * 256;          // N*256 floats (activations)

    // --- CSR of transposed graph + normalization ---
    zero_int_kernel<<<cdiv(N, 256), 256, 0, stream>>>(pos, N);
    count_kernel<<<cdiv(E, 256), 256, 0, stream>>>(dst, pos, E);
    dinv_kernel<<<cdiv(N, 256), 256, 0, stream>>>(pos, dinv, N);
    scan_kernel<<<1, 1024, 0, stream>>>(pos, rowptr, N);
    copy_int_kernel<<<cdiv(N, 256), 256, 0, stream>>>(rowptr, pos, N);
    fill_csr_kernel<<<cdiv(E, 256), 256, 0, stream>>>(src, dst, pos, csr, E);
    sort_buckets_kernel<<<cdiv(N, 256), 256, 0, stream>>>(rowptr, csr, N);

    const int gx = cdiv(N, 64);

    // --- layer 1: 512 -> 256, ReLU ---
    gemm_scale_kernel<64, 8, 2><<<dim3(gx, 4), 256, 0, stream>>>(x, W1, dinv, bufH, N, 512, 256);
    agg_csr_kernel<true, 256><<<cdiv((long long)N * 64, 256), 256, 0, stream>>>(
        bufH, rowptr, csr, dinv, b1, bufG, N);

    // --- layer 2: 256 -> 32, ReLU ---
    gemm_scale_kernel<32, 8, 1><<<dim3(gx, 1), 256, 0, stream>>>(bufG, W2, dinv, bufH, N, 256, 32);
    agg_csr_kernel<true, 32><<<cdiv((long long)N * 8, 256), 256, 0, stream>>>(
        bufH, rowptr, csr, dinv, b2, bufG, N);

    // --- layer 3: 32 -> 16, no activation, output to d_out ---
    gemm_scale_kernel<16, 4, 1><<<dim3(gx, 1), 128, 0, stream>>>(bufG, W3, dinv, bufH, N, 32, 16);
    agg_csr_kernel<false, 16><<<cdiv((long long)N * 4, 256), 256, 0, stream>>>(
        bufH, rowptr, csr, dinv, b3, (float*)d_out, N);
}